// ProbabilisticLabelTree_88579405513417
// MI455X (gfx1250) — compile-verified
//
#include <hip/hip_runtime.h>
#include <hip/hip_bf16.h>
#include <math.h>

#define BATCH   64
#define DIM     768
#define L1N     8192
#define BRANCH  32
#define TOPK    32
#define NCAND   (TOPK * BRANCH)   // 1024

typedef __attribute__((ext_vector_type(2))) float v2f;
typedef __attribute__((ext_vector_type(8))) float v8f;

// sigmoid via v_exp_f32 + hardware v_rcp_f32 (no div_scale/Newton chain)
__device__ __forceinline__ float sigmoidf_fast(float v) {
    return __builtin_amdgcn_rcpf(1.0f + __expf(-v));
}

// ---------------------------------------------------------------------------
// Stage 1: probs0[b, g] = sigmoid(x[b,:] . W0[g,:] + b0[g])
// One wave (32 threads) computes a 16x16 output tile with V_WMMA_F32_16X16X4_F32,
// looping K = 768 in steps of 4. Memory-light (25 MB of W0 total).
// ---------------------------------------------------------------------------
__global__ __launch_bounds__(32)
void plt_stage1_wmma(const float* __restrict__ x,
                     const float* __restrict__ W0,
                     const float* __restrict__ b0,
                     float* __restrict__ probs0) {
    const int n0   = blockIdx.x * 16;          // group tile base
    const int m0   = blockIdx.y * 16;          // batch tile base
    const int lane = threadIdx.x;              // 0..31 (wave32)
    const int half = lane >> 4;                // 0: K+{0,1}, 1: K+{2,3}
    const int l    = lane & 15;

    const float* __restrict__ xrow = x  + (size_t)(m0 + l) * DIM;
    const float* __restrict__ wrow = W0 + (size_t)(n0 + l) * DIM;

    v8f acc = {};
    #pragma unroll 4
    for (int k = 0; k < DIM; k += 4) {
        const int kk = k + half * 2;
        v2f a, b;
        a.x = xrow[kk];     a.y = xrow[kk + 1];   // A: 16x4 fragment (x tile)
        b.x = wrow[kk];     b.y = wrow[kk + 1];   // B: 4x16 fragment (W0^T tile)
        acc = __builtin_amdgcn_wmma_f32_16x16x4_f32(
                  false, a, false, b, (short)0, acc, false, false);
    }

    const int   col  = n0 + l;
    const float bias = b0[col];
    const int   rbase = m0 + half * 8;
    #pragma unroll
    for (int r = 0; r < 8; ++r) {
        float v = sigmoidf_fast(acc[r] + bias);
        probs0[(size_t)(rbase + r) * L1N + col] = v;
    }
}

// ---------------------------------------------------------------------------
// Top-k: per batch row, select 32 largest of 8192 sigmoid scores, emit the
// selected group indices sorted ascending. One wave per row; scores staged in
// LDS (32 KB of the 320 KB WGP pool). 32 argmax sweeps, each a strided scan +
// shfl_xor wave32 reduction.
// ---------------------------------------------------------------------------
__global__ __launch_bounds__(32)
void plt_topk(const float* __restrict__ probs0, int* __restrict__ sel) {
    __shared__ float sv[L1N];
    const int b    = blockIdx.x;
    const int lane = threadIdx.x;

    const float* __restrict__ row = probs0 + (size_t)b * L1N;
    for (int i = lane; i < L1N; i += 32) sv[i] = row[i];
    __syncthreads();

    int mine = 0;   // lane t ends up holding the t-th largest's index
    for (int t = 0; t < TOPK; ++t) {
        float bestv = -3.4e38f;
        int   besti = 0x7fffffff;
        for (int i = lane; i < L1N; i += 32) {
            float v = sv[i];
            if (v > bestv || (v == bestv && i < besti)) { bestv = v; besti = i; }
        }
        #pragma unroll
        for (int off = 16; off > 0; off >>= 1) {
            float ov = __shfl_xor(bestv, off, 32);
            int   oi = __shfl_xor(besti, off, 32);
            if (ov > bestv || (ov == bestv && oi < besti)) { bestv = ov; besti = oi; }
        }
        if (lane == t) mine = besti;
        if (lane == 0) sv[besti] = -3.4e38f;
        __syncthreads();
    }

    // sort ascending by index: rank = #selected indices smaller than mine
    int rank = 0;
    #pragma unroll
    for (int j = 0; j < TOPK; ++j) {
        int oj = __shfl(mine, j, 32);
        rank += (oj < mine) ? 1 : 0;
    }
    sel[b * TOPK + rank] = mine;
}

// ---------------------------------------------------------------------------
// Stage 2: bandwidth-bound gathered GEMV. Each wave scores one (b, candidate):
// 768-dim dot as 6 float4 (b128) loads per lane, coalesced 512 B per wave
// issue, then a 5-step shfl_xor reduction. ~201 MB of W1 traffic dominates
// (~8.6 us at 23.3 TB/s) -> pure bandwidth problem, VALU FMAs suffice.
// ---------------------------------------------------------------------------
__global__ __launch_bounds__(256)
void plt_stage2(const float* __restrict__ x,
                const float* __restrict__ W1,
                const float* __restrict__ b1,
                const float* __restrict__ probs0,
                const int*   __restrict__ sel,
                float* __restrict__ out_probs,
                float* __restrict__ out_cand,
                float* __restrict__ out_mask) {
    const int b    = blockIdx.y;
    const int wave = threadIdx.x >> 5;
    const int lane = threadIdx.x & 31;
    const int c    = blockIdx.x * 8 + wave;        // 0..1023 candidate slot

    const int parent = sel[b * TOPK + (c >> 5)];   // group index
    const int leaf   = parent * BRANCH + (c & 31); // leaf label

    const float4* __restrict__ wr = (const float4*)(W1 + (size_t)leaf * DIM);
    const float4* __restrict__ xr = (const float4*)(x  + (size_t)b    * DIM);

    float acc = 0.0f;
    #pragma unroll
    for (int i = 0; i < DIM / (4 * 32); ++i) {     // 6 iterations
        float4 wv = wr[lane + 32 * i];
        float4 xv = xr[lane + 32 * i];
        acc = fmaf(wv.x, xv.x, acc);
        acc = fmaf(wv.y, xv.y, acc);
        acc = fmaf(wv.z, xv.z, acc);
        acc = fmaf(wv.w, xv.w, acc);
    }
    #pragma unroll
    for (int off = 16; off > 0; off >>= 1)
        acc += __shfl_xor(acc, off, 32);

    if (lane == 0) {
        const float p0 = probs0[(size_t)b * L1N + parent];
        const float p  = p0 * sigmoidf_fast(acc + b1[leaf]);
        const int   o  = b * NCAND + c;
        out_probs[o] = p;
        out_cand[o]  = (float)leaf;
        out_mask[o]  = 1.0f;
    }
}

// ---------------------------------------------------------------------------
extern "C" void kernel_launch(void* const* d_in, const int* in_sizes, int n_in,
                              void* d_out, int out_size, void* d_ws, size_t ws_size,
                              hipStream_t stream) {
    const float* x  = (const float*)d_in[0];
    const float* W0 = (const float*)d_in[1];
    const float* b0 = (const float*)d_in[2];
    const float* W1 = (const float*)d_in[3];
    const float* b1 = (const float*)d_in[4];
    (void)in_sizes; (void)n_in; (void)out_size; (void)ws_size;

    // workspace: probs0 [64*8192] f32 (2 MB), then sel [64*32] i32
    float* probs0 = (float*)d_ws;
    int*   sel    = (int*)((char*)d_ws + (size_t)BATCH * L1N * sizeof(float));

    float* out_probs = (float*)d_out;
    float* out_cand  = out_probs + BATCH * NCAND;
    float* out_mask  = out_cand  + BATCH * NCAND;

    // Stage 1: 512 x 4 tiles of 16x16, one wave each (WMMA f32 16x16x4)
    plt_stage1_wmma<<<dim3(L1N / 16, BATCH / 16), 32, 0, stream>>>(x, W0, b0, probs0);

    // Top-k per batch row
    plt_topk<<<dim3(BATCH), 32, 0, stream>>>(probs0, sel);

    // Stage 2: 64 rows x 128 blocks x 8 waves = 65536 candidate dots
    plt_stage2<<<dim3(NCAND / 8, BATCH), 256, 0, stream>>>(
        x, W1, b1, probs0, sel, out_probs, out_cand, out_mask);
}